// GAT_56822417326654
// MI455X (gfx1250) — compile-verified
//
#include <hip/hip_runtime.h>
#include <hip/hip_bf16.h>
#include <math.h>

#define NEG_SLOPE 0.2f
#define IN_FEATS 256

typedef float v2f __attribute__((ext_vector_type(2)));
typedef float v8f __attribute__((ext_vector_type(8)));

// ---------------- ordered-int float max helpers ----------------
__device__ __forceinline__ int f2oi(float f) {
    int i = __float_as_int(f);
    return (i >= 0) ? i : (i ^ 0x7FFFFFFF);
}
__device__ __forceinline__ float oi2f(int i) {
    return __int_as_float((i >= 0) ? i : (i ^ 0x7FFFFFFF));
}
#define ORDERED_NEG_INF 0x807FFFFF  // f2oi(-INFINITY)

// ---------------- fills ----------------
__global__ void fill_f32(float* __restrict__ p, float v, long long n) {
    long long i = (long long)blockIdx.x * blockDim.x + threadIdx.x;
    if (i < n) p[i] = v;
}
__global__ void fill_i32(int* __restrict__ p, int v, long long n) {
    long long i = (long long)blockIdx.x * blockDim.x + threadIdx.x;
    if (i < n) p[i] = v;
}

// ---------------- fp32 WMMA GEMM: C[M,Nout] = A[M,K] * B[K,Nout] ----------------
// One wave computes one 16x16 tile with V_WMMA_F32_16X16X4_F32.
// A fragment (32-bit 16x4 ISA layout): lanes 0-15 hold {K=k0,k0+1} of rows 0-15,
// lanes 16-31 hold {K=k0+2,k0+3}. B fragment mirrors it (4x16, row striped).
__global__ void wmma_gemm_f32(const float* __restrict__ A,
                              const float* __restrict__ B,
                              float* __restrict__ C,
                              int M, int K, int Nout) {
    int wave = (int)((blockIdx.x * blockDim.x + threadIdx.x) >> 5);
    int lane = threadIdx.x & 31;
    int ntn  = Nout >> 4;
    int tm   = (wave / ntn) << 4;
    int tn   = (wave % ntn) << 4;
    if (tm >= M) return;                 // wave-uniform: EXEC stays all-ones

    int l15 = lane & 15;
    int hi  = lane >> 4;                 // 0: K pair {k0,k0+1}; 1: {k0+2,k0+3}

    v8f acc = {0.f, 0.f, 0.f, 0.f, 0.f, 0.f, 0.f, 0.f};
    const float* arow = A + (size_t)(tm + l15) * K;
    const float* bcol = B + (size_t)tn + l15;

    #pragma unroll 4
    for (int k0 = 0; k0 < K; k0 += 4) {
        int ka = k0 + (hi << 1);
        v2f afrag, bfrag;
        afrag.x = arow[ka];
        afrag.y = arow[ka + 1];
        bfrag.x = bcol[(size_t)ka * Nout];
        bfrag.y = bcol[(size_t)(ka + 1) * Nout];
        acc = __builtin_amdgcn_wmma_f32_16x16x4_f32(
            /*neg_a=*/false, afrag, /*neg_b=*/false, bfrag,
            /*c_mod=*/(short)0, acc, /*reuse_a=*/false, /*reuse_b=*/false);
    }

    // C/D layout: VGPR i -> row (tm + hi*8 + i), col (tn + l15)
    float* crow = C + (size_t)(tm + (hi << 3)) * Nout + tn + l15;
    #pragma unroll
    for (int i = 0; i < 8; ++i)
        crow[(size_t)i * Nout] = acc[i];
}

// ---------------- el/er = sum_o f[n,h,o] * al/ar[h,o] ----------------
__global__ void attn_scores(const float* __restrict__ f,
                            const float* __restrict__ al,
                            const float* __restrict__ ar,
                            float* __restrict__ el, float* __restrict__ er,
                            int N, int H, int O) {
    int idx = blockIdx.x * blockDim.x + threadIdx.x;   // n*H + h
    if (idx >= N * H) return;
    int h = idx % H;
    const float* fp  = f + (size_t)(idx / H) * H * O + (size_t)h * O;
    const float* alp = al + (size_t)h * O;
    const float* arp = ar + (size_t)h * O;
    float sl = 0.f, sr = 0.f;
    for (int o = 0; o < O; ++o) {
        float v = fp[o];
        sl = fmaf(v, alp[o], sl);
        sr = fmaf(v, arp[o], sr);
    }
    el[idx] = sl;
    er[idx] = sr;
}

// ---------------- per-edge logits + segment max over dst ----------------
__global__ void edge_logits_max(const float* __restrict__ el,
                                const float* __restrict__ er,
                                const int* __restrict__ src,
                                const int* __restrict__ dst,
                                float* __restrict__ e_out,
                                int* __restrict__ mOrd,
                                int E, int H) {
    int idx = blockIdx.x * blockDim.x + threadIdx.x;   // e*H + h
    if (idx >= E * H) return;
    int e = idx / H, h = idx - e * H;
    float v = el[src[e] * H + h] + er[dst[e] * H + h];
    v = (v > 0.f) ? v : NEG_SLOPE * v;                  // leaky_relu
    e_out[idx] = v;
    atomicMax(&mOrd[dst[e] * H + h], f2oi(v));          // GLOBAL_ATOMIC_MAX_I32
}

// ---------------- ex = exp(e - m[dst]); segment sum over dst ----------------
__global__ void edge_exp_sum(float* __restrict__ ebuf,
                             const int* __restrict__ dst,
                             const int* __restrict__ mOrd,
                             float* __restrict__ sum,
                             int E, int H) {
    int idx = blockIdx.x * blockDim.x + threadIdx.x;
    if (idx >= E * H) return;
    int e = idx / H, h = idx - e * H;
    int d = dst[e] * H + h;
    float x = expf(ebuf[idx] - oi2f(mOrd[d]));
    ebuf[idx] = x;
    atomicAdd(&sum[d], x);                              // GLOBAL_ATOMIC_ADD_F32
}

// ---------------- a = ex / s[dst] ----------------
__global__ void edge_norm(float* __restrict__ ebuf,
                          const int* __restrict__ dst,
                          const float* __restrict__ sum,
                          int E, int H) {
    int idx = blockIdx.x * blockDim.x + threadIdx.x;
    if (idx >= E * H) return;
    int e = idx / H, h = idx - e * H;
    ebuf[idx] = ebuf[idx] / sum[dst[e] * H + h];
}

// ---------------- agg[dst] += a * f[src]; one block per edge, blockDim = H*O ----------------
__global__ void edge_aggregate(const float* __restrict__ f,
                               const float* __restrict__ attn,
                               const int* __restrict__ src,
                               const int* __restrict__ dst,
                               float* __restrict__ agg,
                               int H, int O) {
    int e = blockIdx.x;
    int c = threadIdx.x;            // 0 .. H*O-1
    int h = c / O;
    int HO = H * O;
    float a = attn[(size_t)e * H + h];
    float v = a * f[(size_t)src[e] * HO + c];
    atomicAdd(&agg[(size_t)dst[e] * HO + c], v);
}

// ---------------- hout = relu(agg + b) ----------------
__global__ void bias_relu(const float* __restrict__ agg,
                          const float* __restrict__ b,
                          float* __restrict__ hout,
                          int N, int HO) {
    long long idx = (long long)blockIdx.x * blockDim.x + threadIdx.x;
    if (idx >= (long long)N * HO) return;
    int c = (int)(idx % HO);
    float v = agg[idx] + b[c];
    hout[idx] = v > 0.f ? v : 0.f;
}

// ---------------- out[n,c] = mean_h (agg[n,h,c] + b[h,c]) ----------------
__global__ void final_mean(const float* __restrict__ agg,
                           const float* __restrict__ b,
                           float* __restrict__ out,
                           int N, int H, int Cc) {
    int idx = blockIdx.x * blockDim.x + threadIdx.x;    // n*Cc + c
    if (idx >= N * Cc) return;
    int n = idx / Cc, c = idx - n * Cc;
    float s = 0.f;
    for (int h = 0; h < H; ++h)
        s += agg[(size_t)n * H * Cc + (size_t)h * Cc + c] + b[h * Cc + c];
    out[idx] = s * (1.f / (float)H);
}

// ---------------- one GAT layer ----------------
static void run_layer(const float* hin, int K, const float* W,
                      const float* al, const float* ar,
                      int H, int O,
                      const int* src, const int* dst, int N, int E,
                      float* f, float* el, float* er, int* mOrd, float* sum,
                      float* ex, float* agg, hipStream_t stream) {
    int HO = H * O;
    // 1) projection GEMM on the matrix pipe
    {
        long long waves = (long long)(N / 16) * (HO / 16);
        long long threads = waves * 32;
        int blocks = (int)((threads + 127) / 128);
        wmma_gemm_f32<<<blocks, 128, 0, stream>>>(hin, W, f, N, K, HO);
    }
    // 2) attention scores
    attn_scores<<<(N * H + 255) / 256, 256, 0, stream>>>(f, al, ar, el, er, N, H, O);
    // 3) init segment buffers (must re-init each call: deterministic launch)
    fill_i32<<<(N * H + 255) / 256, 256, 0, stream>>>(mOrd, (int)ORDERED_NEG_INF, (long long)N * H);
    fill_f32<<<(N * H + 255) / 256, 256, 0, stream>>>(sum, 0.f, (long long)N * H);
    fill_f32<<<(int)(((long long)N * HO + 255) / 256), 256, 0, stream>>>(agg, 0.f, (long long)N * HO);
    // 4) edge softmax (max -> exp/sum -> normalize)
    int egrid = (E * H + 255) / 256;
    edge_logits_max<<<egrid, 256, 0, stream>>>(el, er, src, dst, ex, mOrd, E, H);
    edge_exp_sum  <<<egrid, 256, 0, stream>>>(ex, dst, mOrd, sum, E, H);
    edge_norm     <<<egrid, 256, 0, stream>>>(ex, dst, sum, E, H);
    // 5) weighted scatter-add of messages (coalesced row per edge, L2 atomics)
    edge_aggregate<<<E, HO, 0, stream>>>(f, ex, src, dst, agg, H, O);
}

extern "C" void kernel_launch(void* const* d_in, const int* in_sizes, int n_in,
                              void* d_out, int out_size, void* d_ws, size_t ws_size,
                              hipStream_t stream) {
    const float* x   = (const float*)d_in[0];
    const float* W0  = (const float*)d_in[1];
    const float* al0 = (const float*)d_in[2];
    const float* ar0 = (const float*)d_in[3];
    const float* b0  = (const float*)d_in[4];
    const float* W1  = (const float*)d_in[5];
    const float* al1 = (const float*)d_in[6];
    const float* ar1 = (const float*)d_in[7];
    const float* b1  = (const float*)d_in[8];
    const float* W2  = (const float*)d_in[9];
    const float* al2 = (const float*)d_in[10];
    const float* ar2 = (const float*)d_in[11];
    const float* b2  = (const float*)d_in[12];
    const int*   src = (const int*)d_in[13];
    const int*   dst = (const int*)d_in[14];

    const int N = in_sizes[0] / IN_FEATS;   // 50000
    const int E = in_sizes[13];             // 800000
    const int H = 4;
    const int HID = 256;                    // H * 64
    const int C = 40;

    // carve workspace (all L2-resident working set)
    char* w = (char*)d_ws;
    float* f    = (float*)w;  w += (size_t)N * HID * sizeof(float);
    float* agg  = (float*)w;  w += (size_t)N * HID * sizeof(float);
    float* hbuf = (float*)w;  w += (size_t)N * HID * sizeof(float);
    float* el   = (float*)w;  w += (size_t)N * H * sizeof(float);
    float* er   = (float*)w;  w += (size_t)N * H * sizeof(float);
    int*   mOrd = (int*)w;    w += (size_t)N * H * sizeof(int);
    float* sum  = (float*)w;  w += (size_t)N * H * sizeof(float);
    float* ex   = (float*)w;  w += (size_t)E * H * sizeof(float);
    (void)ws_size; (void)n_in; (void)out_size;

    // ---- layer 0: [N,256] -> [N,4,64] ----
    run_layer(x, IN_FEATS, W0, al0, ar0, H, 64, src, dst, N, E,
              f, el, er, mOrd, sum, ex, agg, stream);
    bias_relu<<<(int)(((long long)N * HID + 255) / 256), 256, 0, stream>>>(agg, b0, hbuf, N, HID);

    // ---- layer 1: [N,256] -> [N,4,64] ----
    run_layer(hbuf, HID, W1, al1, ar1, H, 64, src, dst, N, E,
              f, el, er, mOrd, sum, ex, agg, stream);
    bias_relu<<<(int)(((long long)N * HID + 255) / 256), 256, 0, stream>>>(agg, b1, hbuf, N, HID);

    // ---- layer 2: [N,256] -> [N,4,40], mean over heads ----
    run_layer(hbuf, HID, W2, al2, ar2, H, C, src, dst, N, E,
              f, el, er, mOrd, sum, ex, agg, stream);
    final_mean<<<(N * C + 255) / 256, 256, 0, stream>>>(agg, b2, (float*)d_out, N, H, C);
}